// TransformLayer_33921651704133
// MI455X (gfx1250) — compile-verified
//
#include <hip/hip_runtime.h>
#include <math.h>
#include <stdint.h>

// Problem constants (fixed by the reference).
#define NROWS  131072
#define NCOLS  64
#define NREP   8
#define OUTC   (NCOLS * NREP)   // 512
#define NBINS  65536            // top-16-bit radix buckets
#define NELEM  (NROWS * NCOLS)  // 8,388,608
#define BINS_PER_T (NBINS / 256)

typedef unsigned int       u32;
typedef unsigned long long u64;

// Workspace layout (total 96 MB; fits MI455X's 192MB L2 together with y):
//   [0,   16MB) : hist  u32[64][65536]  (counts; consumed by scatter)
//   [16,  32MB) : base  u32[64][65536]  (exclusive-scanned bucket starts)
//   [32,  96MB) : scat  u64[64][131072] ((low16<<32)|row, bucket-grouped)
#define HIST_BYTES ((size_t)NCOLS * NBINS * sizeof(u32))
#define BASE_BYTES ((size_t)NCOLS * NBINS * sizeof(u32))

#if defined(__AMDGCN__) && __has_builtin(__builtin_amdgcn_wmma_f32_16x16x4_f32)
#define USE_WMMA_SCAN 1
#else
#define USE_WMMA_SCAN 0
#endif

// Order-preserving float -> uint transform (branchless).
__device__ __forceinline__ u32 fl2u(float f) {
    u32 u = __float_as_uint(f);
    u32 m = (u32)(-(int)(u >> 31)) | 0x80000000u;
    return u ^ m;
}

__device__ __forceinline__ float softplusf(float x) {
    return fmaxf(x, 0.0f) + log1pf(expf(-fabsf(x)));
}

// ---------------------------------------------------------------- zero hist
__global__ void k_zero(u32* __restrict__ p, int n) {
    int i = blockIdx.x * blockDim.x + threadIdx.x;
    if (i < n) p[i] = 0u;
}

// ------------------------------------------------- per-column 64K histogram
// Stages the tile through LDS with the CDNA5 async global->LDS path
// (ASYNCcnt-tracked), then histograms with global atomics into L2.
__global__ void k_hist(const float* __restrict__ y, u32* __restrict__ hist) {
    __shared__ float sy[256];
    int t = blockIdx.x * blockDim.x + threadIdx.x;
    float yv;
#if defined(__AMDGCN__)
    {
        // Low 32 bits of the generic pointer to a __shared__ object are the
        // LDS byte address (aperture mapping truncates to addr[31:0]).
        unsigned lds_addr    = (unsigned)(uintptr_t)&sy[threadIdx.x];
        unsigned long long g = (unsigned long long)(uintptr_t)(y + t);
        asm volatile("global_load_async_to_lds_b32 %0, %1, off"
                     :: "v"(lds_addr), "v"(g) : "memory");
        asm volatile("s_wait_asynccnt 0x0" ::: "memory");
        yv = sy[threadIdx.x];   // each lane reads only its own slot
    }
#else
    yv = y[t];
#endif
    u32 u = fl2u(yv);
    int c = t & (NCOLS - 1);
    atomicAdd(&hist[(size_t)c * NBINS + (u >> 16)], 1u);
}

#if USE_WMMA_SCAN
typedef float v2f __attribute__((ext_vector_type(2)));
typedef float v8f __attribute__((ext_vector_type(8)));

// Inclusive scan of 256 f32 values (exact integers < 2^24) held in ts[256],
// executed by ONE full wave32 (EXEC all ones), result into ss[256].
// Row-major 16x16 matrix X;  scan(X) = X*U + (Ls*X)*J  via 12 chained
// v_wmma_f32_16x16x4_f32 ops (U upper-tri ones, Ls strict lower-tri, J ones).
__device__ __forceinline__ void wave_scan256_wmma(const float* ts, float* ss,
                                                  float* yld, int l) {
    const int n     = l & 15;    // C/D + B column for this lane
    const int m     = l & 15;    // A row for this lane
    const int khalf = l >> 4;    // lanes 16-31 carry the upper K pair

    // ---- acc = X * U
    v8f acc = {};
#pragma unroll
    for (int q = 0; q < 4; ++q) {
        const int k0 = 4 * q + 2 * khalf;
        v2f a; a.x = ts[m * 16 + k0]; a.y = ts[m * 16 + k0 + 1];
        v2f b; b.x = (k0     <= n) ? 1.0f : 0.0f;
               b.y = (k0 + 1 <= n) ? 1.0f : 0.0f;
        acc = __builtin_amdgcn_wmma_f32_16x16x4_f32(false, a, false, b,
                                                    (short)0, acc, false, false);
    }
    // ---- Y = Ls * X
    v8f yv = {};
#pragma unroll
    for (int q = 0; q < 4; ++q) {
        const int k0 = 4 * q + 2 * khalf;
        v2f a; a.x = (k0     < m) ? 1.0f : 0.0f;
               a.y = (k0 + 1 < m) ? 1.0f : 0.0f;
        v2f b; b.x = ts[(k0    ) * 16 + n];
               b.y = ts[(k0 + 1) * 16 + n];
        yv = __builtin_amdgcn_wmma_f32_16x16x4_f32(false, a, false, b,
                                                   (short)0, yv, false, false);
    }
    // Spill Y (C layout) so it can be re-read in A layout (cross-lane move).
#pragma unroll
    for (int i = 0; i < 8; ++i) yld[(i + 8 * khalf) * 16 + n] = yv[i];
    __threadfence_block();
    // ---- acc += Y * J
#pragma unroll
    for (int q = 0; q < 4; ++q) {
        const int k0 = 4 * q + 2 * khalf;
        v2f a; a.x = yld[m * 16 + k0]; a.y = yld[m * 16 + k0 + 1];
        v2f b; b.x = 1.0f; b.y = 1.0f;
        acc = __builtin_amdgcn_wmma_f32_16x16x4_f32(false, a, false, b,
                                                    (short)0, acc, false, false);
    }
#pragma unroll
    for (int i = 0; i < 8; ++i) ss[(i + 8 * khalf) * 16 + n] = acc[i];
}
#endif // USE_WMMA_SCAN

// ------------------------------------- exclusive scan of 65536 bins / column
// One 256-thread workgroup per column; each thread owns 256 consecutive bins.
// The 256 thread-sums are scanned by wave 0 with WMMA (fallback: LDS scan).
__global__ void k_scan(const u32* __restrict__ hist, u32* __restrict__ base) {
    __shared__ float ts[256];
    __shared__ float ss[256];
#if USE_WMMA_SCAN
    __shared__ float yld[256];
#endif
    int c = blockIdx.x;
    int t = threadIdx.x;
    const u32* h = hist + (size_t)c * NBINS;
    u32*       b = base + (size_t)c * NBINS;
    const int i0 = t * BINS_PER_T;

    u32 sum = 0;
    for (int i = 0; i < BINS_PER_T; ++i) sum += h[i0 + i];
    ts[t] = (float)sum;                       // exact: sum <= 131072 < 2^24
    __syncthreads();

#if USE_WMMA_SCAN
    if (t < 32) wave_scan256_wmma(ts, ss, yld, t);   // wave 0, EXEC all ones
#else
    ss[t] = ts[t];
    __syncthreads();
    for (int off = 1; off < 256; off <<= 1) {
        float v = (t >= off) ? ss[t - off] : 0.0f;
        __syncthreads();
        ss[t] += v;
        __syncthreads();
    }
#endif
    __syncthreads();

    u32 run = (t == 0) ? 0u : (u32)ss[t - 1];        // exclusive prefix, exact
    for (int i = 0; i < BINS_PER_T; ++i) {
        u32 cb = h[i0 + i];
        b[i0 + i] = run;
        run += cb;
    }
}

// -------------------------------------- scatter (low16,row) into its bucket
// Uses hist as a down-counting cursor: old in {cnt..1}  ->  pos in [b0,b1).
__global__ void k_scatter(const float* __restrict__ y,
                          u32* __restrict__ hist,
                          const u32* __restrict__ base,
                          u64* __restrict__ scat) {
    int t = blockIdx.x * blockDim.x + threadIdx.x;
    u32 u   = fl2u(y[t]);
    int c   = t & (NCOLS - 1);
    u32 row = (u32)t >> 6;
    u32 H = u >> 16, L = u & 0xFFFFu;
    u32 old = atomicSub(&hist[(size_t)c * NBINS + H], 1u);
    u32 nb  = (H == NBINS - 1) ? (u32)NROWS : base[(size_t)c * NBINS + H + 1];
    u32 pos = nb - old;
    scat[(size_t)c * NROWS + pos] = ((u64)L << 32) | (u64)row;
}

// ------------------- rank via in-bucket counting + transform + tiled store
__global__ void k_final(const float* __restrict__ y,
                        const u32* __restrict__ base,
                        const u64* __restrict__ scat,
                        float* __restrict__ out,
                        const float* __restrict__ pA,
                        const float* __restrict__ pB,
                        const float* __restrict__ pg,
                        const float* __restrict__ pk) {
    int t = blockIdx.x * blockDim.x + threadIdx.x;   // coalesced over y & out
    int c   = t & (NCOLS - 1);
    u32 row = (u32)t >> 6;

    u32 u = fl2u(y[t]);
    u32 H = u >> 16, L = u & 0xFFFFu;
    const u32* bc = base + (size_t)c * NBINS;
    u32 b0 = bc[H];
    u32 b1 = (H == NBINS - 1) ? (u32)NROWS : bc[H + 1];

    const u64* sp = scat + (size_t)c * NROWS;
    __builtin_prefetch(sp + b0, 0, 1);               // global_prefetch_b8

    u64 my = ((u64)L << 32) | (u64)row;              // stable tiebreak by row
    u32 lt = 0;
    for (u32 p = b0; p < b1; ++p) lt += (sp[p] < my) ? 1u : 0u;
    u32 rank0 = b0 + lt;                             // 0-based global rank

    // cdf = (rank1 + 1) / (N + 2), rank1 = rank0 + 1
    float cdf = (float)(rank0 + 2) * (1.0f / (float)(NROWS + 2));
    float z = 1.41421356237309515f * erfinvf(fmaf(2.0f, cdf, -1.0f));

    float A = pA[0];
    float B = softplusf(pB[0]);
    float g = pg[0];
    float k = softplusf(pk[0]) - 0.5f;

    float w = tanhf(0.5f * g * z);
    float v = z * powf(fmaf(z, z, 1.0f), k);
    float x = fmaf(B * fmaf(0.8f, w, 1.0f), v, A);

    size_t ob = (size_t)row * OUTC + (size_t)c;
#pragma unroll
    for (int r = 0; r < NREP; ++r) out[ob + (size_t)r * NCOLS] = x;
}

extern "C" void kernel_launch(void* const* d_in, const int* in_sizes, int n_in,
                              void* d_out, int out_size, void* d_ws, size_t ws_size,
                              hipStream_t stream) {
    (void)in_sizes; (void)n_in; (void)out_size; (void)ws_size;

    const float* y  = (const float*)d_in[0];
    const float* A_ = (const float*)d_in[1];
    const float* B_ = (const float*)d_in[2];
    const float* g_ = (const float*)d_in[3];
    const float* k_ = (const float*)d_in[4];
    float* out = (float*)d_out;

    u32* hist = (u32*)d_ws;
    u32* base = (u32*)((char*)d_ws + HIST_BYTES);
    u64* scat = (u64*)((char*)d_ws + HIST_BYTES + BASE_BYTES);

    const int nh   = NCOLS * NBINS;         // 4M bins
    const int nblk = NELEM / 256;           // 32768 blocks of 256 (8 waves)

    k_zero   <<<(nh + 255) / 256, 256, 0, stream>>>(hist, nh);
    k_hist   <<<nblk, 256, 0, stream>>>(y, hist);
    k_scan   <<<NCOLS, 256, 0, stream>>>(hist, base);
    k_scatter<<<nblk, 256, 0, stream>>>(y, hist, base, scat);
    k_final  <<<nblk, 256, 0, stream>>>(y, base, scat, out, A_, B_, g_, k_);
}